// NodeAttention_88218628260359
// MI455X (gfx1250) — compile-verified
//
#include <hip/hip_runtime.h>

typedef __attribute__((ext_vector_type(2))) float v2f;
typedef __attribute__((ext_vector_type(8))) float v8f;

#define NN 50000
#define EE 800000
#define DN 64
#define DE 32
#define HH 4
#define DH 16

// Monotone float <-> uint encoding for atomic max on signed floats.
__device__ __forceinline__ unsigned fenc(float f) {
  unsigned u = __float_as_uint(f);
  return (u & 0x80000000u) ? ~u : (u | 0x80000000u);
}
__device__ __forceinline__ float fdec(unsigned e) {
  unsigned u = (e & 0x80000000u) ? (e & 0x7FFFFFFFu) : ~e;
  return __uint_as_float(u);
}

// ---------------- kernel 0: init accumulators ----------------
__global__ void init_k(float* __restrict__ acc, float* __restrict__ denom,
                       unsigned* __restrict__ mxe) {
  int i = blockIdx.x * blockDim.x + threadIdx.x;   // 0 .. N*64-1
  acc[i] = 0.0f;
  if (i < NN * HH) {
    denom[i] = 0.0f;
    mxe[i] = fenc(-1e9f);
  }
}

// ---------------- kernel 1: fused Q/K/V projection via WMMA f32 ----------------
// block = 128 threads (4 waves); block b -> rows [16b,16b+16); wave w -> cols [16w,16w+16)
__global__ void qkv_wmma(const float* __restrict__ X,
                         const float* __restrict__ Wq, const float* __restrict__ bq,
                         const float* __restrict__ Wk, const float* __restrict__ bk,
                         const float* __restrict__ Wv, const float* __restrict__ bv,
                         float* __restrict__ Q, float* __restrict__ K,
                         float* __restrict__ V) {
  const int lane = threadIdx.x & 31;
  const int wave = threadIdx.x >> 5;
  const int row0 = blockIdx.x * 16;
  const int col0 = wave * 16;
  const int m  = lane & 15;   // M (for A) / N (for B)
  const int kh = lane >> 4;   // which K-pair this half-wave holds

  const float2* X2  = (const float2*)X;
  const float2* Wq2 = (const float2*)Wq;
  const float2* Wk2 = (const float2*)Wk;
  const float2* Wv2 = (const float2*)Wv;

  v8f cq = {}; v8f ck = {}; v8f cv = {};
#pragma unroll
  for (int k0 = 0; k0 < DN; k0 += 4) {
    const int kk = k0 + 2 * kh;
    // A frag: X[row0+m][kk], X[row0+m][kk+1]  (shared across Q/K/V chains)
    float2 av = X2[((row0 + m) * DN + kk) >> 1];
    v2f a; a.x = av.x; a.y = av.y;
    // B frag for X @ W^T: B[k][n] = W[n][k]  -> 2 contiguous floats of row (col0+m)
    float2 b0 = Wq2[((col0 + m) * DN + kk) >> 1];
    v2f bqf; bqf.x = b0.x; bqf.y = b0.y;
    cq = __builtin_amdgcn_wmma_f32_16x16x4_f32(false, a, false, bqf, (short)0, cq, false, false);
    float2 b1 = Wk2[((col0 + m) * DN + kk) >> 1];
    v2f bkf; bkf.x = b1.x; bkf.y = b1.y;
    ck = __builtin_amdgcn_wmma_f32_16x16x4_f32(false, a, false, bkf, (short)0, ck, false, false);
    float2 b2 = Wv2[((col0 + m) * DN + kk) >> 1];
    v2f bvf; bvf.x = b2.x; bvf.y = b2.y;
    cv = __builtin_amdgcn_wmma_f32_16x16x4_f32(false, a, false, bvf, (short)0, cv, false, false);
  }
  // C/D layout: VGPR r, lanes 0-15 -> (M=r, N=lane); lanes 16-31 -> (M=r+8, N=lane-16)
#pragma unroll
  for (int r = 0; r < 8; ++r) {
    const int row = row0 + r + 8 * kh;
    const int col = col0 + m;
    Q[row * DN + col] = cq[r] + bq[col];
    K[row * DN + col] = ck[r] + bk[col];
    V[row * DN + col] = cv[r] + bv[col];
  }
}

// ---------------- kernel 2: per-edge scores + segment max ----------------
__global__ void edge_scores(const float* __restrict__ Q, const float* __restrict__ K,
                            const float* __restrict__ EF, const int* __restrict__ EI,
                            const float* __restrict__ We, const float* __restrict__ be,
                            const float* __restrict__ log_temp,
                            float* __restrict__ scores, unsigned* __restrict__ mxe) {
  const int tid = blockIdx.x * blockDim.x + threadIdx.x;  // E*H threads exactly
  const int e = tid >> 2;
  const int h = tid & 3;
  const int s = EI[e];
  const int t = EI[EE + e];

  const float4* q4 = (const float4*)(Q + (size_t)t * DN + h * DH);
  const float4* k4 = (const float4*)(K + (size_t)s * DN + h * DH);
  float dot = 0.0f;
#pragma unroll
  for (int i = 0; i < 4; ++i) {
    float4 qa = q4[i], ka = k4[i];
    dot += qa.x * ka.x + qa.y * ka.y + qa.z * ka.z + qa.w * ka.w;
  }
  dot *= 0.25f;  // 1/sqrt(D_H)

  const float4* e4 = (const float4*)(EF + (size_t)e * DE);
  const float4* w4 = (const float4*)(We + h * DE);
  float ep = 0.0f;
#pragma unroll
  for (int i = 0; i < 8; ++i) {
    float4 ea = e4[i], wa = w4[i];
    ep += ea.x * wa.x + ea.y * wa.y + ea.z * wa.z + ea.w * wa.w;
  }

  const float sc = (dot + ep + be[h]) * expf(log_temp[h]);
  scores[(size_t)e * HH + h] = sc;
  atomicMax(&mxe[t * HH + h], fenc(sc));
}

// ---------------- kernel 3: scores -> exp weights (in place) + denominator ----------------
// E*H threads: computes each distinct exp exactly once (the big scatter pass
// below then does zero transcendental work), and accumulates the softmax
// denominator without any branch divergence.
__global__ void edge_exp(const int* __restrict__ EI, float* __restrict__ scores,
                         const unsigned* __restrict__ mxe, float* __restrict__ denom) {
  const int tid = blockIdx.x * blockDim.x + threadIdx.x;  // E*H threads exactly
  const int e = tid >> 2;
  const int h = tid & 3;
  const int t = EI[EE + e];
  const float ex = expf(scores[tid] - fdec(mxe[t * HH + h]));
  scores[tid] = ex;
  atomicAdd(&denom[t * HH + h], ex);
}

// ---------------- kernel 4: exp-weighted V scatter (pure stream + atomics) ----------------
// thread per (edge, dim): one cached weight load, one V load, one f32 atomic.
__global__ void edge_scatter(const float* __restrict__ V, const int* __restrict__ EI,
                             const float* __restrict__ scores,
                             float* __restrict__ acc) {
  const int tid = blockIdx.x * blockDim.x + threadIdx.x;  // E*64 threads exactly
  const int e = tid >> 6;
  const int d = tid & 63;
  const int h = d >> 4;
  const int s = EI[e];
  const int t = EI[EE + e];
  const float ex = scores[(size_t)e * HH + h];
  atomicAdd(&acc[(size_t)t * DN + d], ex * V[(size_t)s * DN + d]);
}

// ---------------- kernel 5: normalize + output GEMM (WMMA) + residual + LN ----------------
__global__ void out_ln_wmma(const float* __restrict__ X, const float* __restrict__ acc,
                            const float* __restrict__ denom,
                            const float* __restrict__ Wo, const float* __restrict__ bo,
                            const float* __restrict__ gamma, const float* __restrict__ beta,
                            float* __restrict__ out) {
  __shared__ float yt[16][DN + 4];
  const int lane = threadIdx.x & 31;
  const int wave = threadIdx.x >> 5;
  const int row0 = blockIdx.x * 16;
  const int col0 = wave * 16;
  const int m  = lane & 15;
  const int kh = lane >> 4;

  // per-row, per-head reciprocal softmax denominators folded into A fragment
  float rd[4];
#pragma unroll
  for (int j = 0; j < 4; ++j)
    rd[j] = 1.0f / (denom[(row0 + m) * HH + j] + 1e-10f);

  const float2* A2 = (const float2*)acc;
  const float2* W2 = (const float2*)Wo;
  v8f c = {};
#pragma unroll
  for (int k0 = 0; k0 < DN; k0 += 4) {
    const int kk = k0 + 2 * kh;          // even; kk and kk+1 share a head
    float2 av = A2[((row0 + m) * DN + kk) >> 1];
    const float r = rd[kk >> 4];
    v2f a; a.x = av.x * r; a.y = av.y * r;
    float2 bv = W2[((col0 + m) * DN + kk) >> 1];
    v2f b; b.x = bv.x; b.y = bv.y;
    c = __builtin_amdgcn_wmma_f32_16x16x4_f32(false, a, false, b, (short)0, c, false, false);
  }
  // residual into LDS tile
#pragma unroll
  for (int r = 0; r < 8; ++r) {
    const int rl  = r + 8 * kh;
    const int col = col0 + m;
    yt[rl][col] = c[r] + bo[col] + X[(size_t)(row0 + rl) * DN + col];
  }
  __syncthreads();

  // LayerNorm: 8 threads per row, shfl_xor reduction within aligned groups of 8
  const int r  = threadIdx.x >> 3;        // 0..15
  const int c8 = (threadIdx.x & 7) * 8;   // 0..56
  float s = 0.0f, s2 = 0.0f;
#pragma unroll
  for (int j = 0; j < 8; ++j) {
    const float v = yt[r][c8 + j];
    s += v; s2 += v * v;
  }
#pragma unroll
  for (int off = 1; off < 8; off <<= 1) {
    s  += __shfl_xor(s,  off, 32);
    s2 += __shfl_xor(s2, off, 32);
  }
  const float mu   = s * (1.0f / 64.0f);
  const float var  = s2 * (1.0f / 64.0f) - mu * mu;
  const float rstd = rsqrtf(var + 1e-5f);
#pragma unroll
  for (int j = 0; j < 8; ++j) {
    const int cc = c8 + j;
    out[(size_t)(row0 + r) * DN + cc] = (yt[r][cc] - mu) * rstd * gamma[cc] + beta[cc];
  }
}

extern "C" void kernel_launch(void* const* d_in, const int* in_sizes, int n_in,
                              void* d_out, int out_size, void* d_ws, size_t ws_size,
                              hipStream_t stream) {
  const float* X  = (const float*)d_in[0];
  const float* EF = (const float*)d_in[1];
  const int*   EI = (const int*)d_in[2];
  const float* Wq = (const float*)d_in[3];
  const float* bq = (const float*)d_in[4];
  const float* Wk = (const float*)d_in[5];
  const float* bk = (const float*)d_in[6];
  const float* Wv = (const float*)d_in[7];
  const float* bv = (const float*)d_in[8];
  const float* We = (const float*)d_in[9];
  const float* be = (const float*)d_in[10];
  const float* Wo = (const float*)d_in[11];
  const float* bo = (const float*)d_in[12];
  const float* gm = (const float*)d_in[13];
  const float* bt = (const float*)d_in[14];
  const float* lt = (const float*)d_in[15];
  float* out = (float*)d_out;

  // workspace layout (floats): Q,K,V [N*64 each] | scores/exp [E*4] | acc [N*64] |
  // denom [N*4] | mxe [N*4 u32]   => ~65.6 MB total
  float* Q      = (float*)d_ws;
  float* K      = Q + (size_t)NN * DN;
  float* V      = K + (size_t)NN * DN;
  float* scores = V + (size_t)NN * DN;
  float* acc    = scores + (size_t)EE * HH;
  float* denom  = acc + (size_t)NN * DN;
  unsigned* mxe = (unsigned*)(denom + (size_t)NN * HH);

  init_k      <<<(NN * DN) / 256, 256, 0, stream>>>(acc, denom, mxe);
  qkv_wmma    <<<NN / 16, 128, 0, stream>>>(X, Wq, bq, Wk, bk, Wv, bv, Q, K, V);
  edge_scores <<<(EE * HH) / 256, 256, 0, stream>>>(Q, K, EF, EI, We, be, lt, scores, mxe);
  edge_exp    <<<(EE * HH) / 256, 256, 0, stream>>>(EI, scores, mxe, denom);
  edge_scatter<<<(EE * 64) / 256, 256, 0, stream>>>(V, EI, scores, acc);
  out_ln_wmma <<<NN / 16, 128, 0, stream>>>(X, acc, denom, Wo, bo, gm, bt, out);
}